// DeepFwFM_12610023981507
// MI455X (gfx1250) — compile-verified
//
#include <hip/hip_runtime.h>
#include <hip/hip_bf16.h>

typedef float v2f __attribute__((ext_vector_type(2)));
typedef float v8f __attribute__((ext_vector_type(8)));

#define BATCH 16384
#define NF    39
#define NUMF  13
#define CATF  26
#define EMB   10
#define DIN   390   // NF * EMB
#define KP    400   // K padded to multiple of 4 (and equal to hidden width)
#define NH    400   // hidden width

// ---------- pack weights [N, Kin] row-major -> [KP, N] with zero pad ----------
__global__ __launch_bounds__(256) void pack_w_kernel(const float* __restrict__ W,
                                                     float* __restrict__ Wp, int Kin) {
    int i = blockIdx.x * blockDim.x + threadIdx.x;   // over KP*NH
    if (i >= KP * NH) return;
    int k = i / NH, n = i % NH;
    Wp[i] = (k < Kin) ? W[n * Kin + k] : 0.0f;
}

// ---------- build padded V matrix [BATCH, KP] ----------
__global__ __launch_bounds__(256) void build_v_kernel(const int* __restrict__ Xi,
                                                      const float* __restrict__ Xv,
                                                      const float* __restrict__ cat_table,
                                                      const float* __restrict__ num_emb,
                                                      const int* __restrict__ offsets,
                                                      float* __restrict__ Vp) {
    int i = blockIdx.x * blockDim.x + threadIdx.x;   // over BATCH*KP
    if (i >= BATCH * KP) return;
    int b = i / KP, c = i % KP;
    float val = 0.0f;
    if (c < DIN) {
        int f = c / EMB, d = c % EMB;
        if (f < NUMF) {
            val = num_emb[f * EMB + d] * Xv[b * NF + f];
        } else {
            int fi = f - NUMF;
            long idx = (long)Xi[b * CATF + fi] + (long)offsets[fi];
            val = cat_table[idx * (long)EMB + d];
        }
    }
    Vp[i] = val;
}

// ---------- first order + FwFM second order, one wave of 32 rows per block ----------
__global__ __launch_bounds__(32) void fwfm_fs_kernel(const float* __restrict__ Vp,
                                                     const float* __restrict__ fwfm_w,
                                                     const float* __restrict__ FC,
                                                     float* __restrict__ fs) {
    __shared__ float sV[32 * DIN];
    int b0 = blockIdx.x * 32;
    for (int i = threadIdx.x; i < 32 * DIN; i += 32) {
        int r = i / DIN, c = i % DIN;
        sV[i] = Vp[(size_t)(b0 + r) * KP + c];
    }
    __syncthreads();
    const float* v = &sV[threadIdx.x * DIN];
    float first = 0.0f, second = 0.0f;
    for (int k = 0; k < NF; ++k) {
        float vk[EMB];
        #pragma unroll
        for (int d = 0; d < EMB; ++d) vk[d] = v[k * EMB + d];
        #pragma unroll
        for (int d = 0; d < EMB; ++d) first += vk[d] * fwfm_w[k * EMB + d];
        for (int l = k + 1; l < NF; ++l) {
            float s = 0.5f * (FC[k * NF + l] + FC[l * NF + k]);
            float dp = 0.0f;
            #pragma unroll
            for (int d = 0; d < EMB; ++d) dp += vk[d] * v[l * EMB + d];
            second += s * dp;
        }
    }
    fs[b0 + threadIdx.x] = first + second;
}

// ---------- fp32 WMMA GEMM: C[M,NH] = relu(A[M,KP] * Bp[KP,NH] + bias) ----------
// One wave per 16(M) x 80(N) tile; 5120 waves exactly cover M=16384, N=400.
__global__ __launch_bounds__(256) void gemm_relu_wmma_kernel(const float* __restrict__ A,
                                                             const float* __restrict__ Bp,
                                                             const float* __restrict__ bias,
                                                             float* __restrict__ C,
                                                             int do_relu) {
    const int lane  = threadIdx.x & 31;
    const int wid   = (blockIdx.x * blockDim.x + threadIdx.x) >> 5;  // 0..5119
    const int mtile = wid / 5;
    const int n0    = (wid % 5) * 80;
    const int ml    = lane & 15;
    const int hi    = lane >> 4;
    const int m     = mtile * 16 + ml;

    v8f acc[5];
    #pragma unroll
    for (int t = 0; t < 5; ++t) {
        v8f z = {0.f, 0.f, 0.f, 0.f, 0.f, 0.f, 0.f, 0.f};
        acc[t] = z;
    }

    for (int kb = 0; kb < KP; kb += 4) {
        // A fragment (16x4 f32): lane holds row m, k = kb + 2*hi + {0,1}
        v2f a = *(const v2f*)(A + (size_t)m * KP + kb + 2 * hi);
        const float* brow = Bp + (size_t)(kb + 2 * hi) * NH;
        #pragma unroll
        for (int t = 0; t < 5; ++t) {
            int n = n0 + t * 16 + ml;
            // B fragment (4x16 f32): lane holds col n, k = kb + 2*hi + {0,1}
            v2f bb;
            bb.x = brow[n];
            bb.y = brow[NH + n];
            acc[t] = __builtin_amdgcn_wmma_f32_16x16x4_f32(
                false, a, false, bb, (short)0, acc[t], false, false);
        }
    }

    #pragma unroll
    for (int t = 0; t < 5; ++t) {
        int n = n0 + t * 16 + ml;
        float bn = bias[n];
        #pragma unroll
        for (int r = 0; r < 8; ++r) {
            int mo = mtile * 16 + r + 8 * hi;   // C layout: vgpr r -> row r + 8*hi
            float val = acc[t][r] + bn;
            if (do_relu) val = fmaxf(val, 0.0f);
            C[(size_t)mo * NH + n] = val;
        }
    }
}

// ---------- final: out = fs + H2 . W3 + bias ----------
__global__ __launch_bounds__(256) void final_kernel(const float* __restrict__ H2,
                                                    const float* __restrict__ W3,
                                                    const float* __restrict__ fs,
                                                    const float* __restrict__ bias,
                                                    float* __restrict__ out) {
    int b = blockIdx.x * blockDim.x + threadIdx.x;
    if (b >= BATCH) return;
    const float4* h = (const float4*)(H2 + (size_t)b * NH);
    const float4* w = (const float4*)W3;
    float acc = 0.0f;
    #pragma unroll 4
    for (int i = 0; i < NH / 4; ++i) {
        float4 x = h[i], y = w[i];
        acc += x.x * y.x + x.y * y.y + x.z * y.z + x.w * y.w;
    }
    out[b] = fs[b] + acc + bias[0];
}

extern "C" void kernel_launch(void* const* d_in, const int* in_sizes, int n_in,
                              void* d_out, int out_size, void* d_ws, size_t ws_size,
                              hipStream_t stream) {
    const int*   Xi      = (const int*)d_in[0];
    const float* Xv      = (const float*)d_in[1];
    const float* cat     = (const float*)d_in[2];
    const float* num_emb = (const float*)d_in[3];
    const int*   offsets = (const int*)d_in[4];
    const float* fwfm_w  = (const float*)d_in[5];
    const float* fcov    = (const float*)d_in[6];
    const float* W1      = (const float*)d_in[7];
    const float* b1      = (const float*)d_in[8];
    const float* W2      = (const float*)d_in[9];
    const float* b2      = (const float*)d_in[10];
    const float* W3      = (const float*)d_in[11];
    const float* bias    = (const float*)d_in[12];
    float* out = (float*)d_out;

    char* ws = (char*)d_ws;
    float* Vp  = (float*)(ws);                 // 16384*400*4 = 26,214,400 B
    float* H1  = (float*)(ws + 26214400);      // 26,214,400 B
    float* W1p = (float*)(ws + 52428800);      // 640,000 B
    float* W2p = (float*)(ws + 53068800);      // 640,000 B
    float* fsb = (float*)(ws + 53708800);      // 65,536 B  (total ~51.3 MB)
    float* H2  = Vp;                           // alias: Vp dead after layer 1

    pack_w_kernel<<<(KP * NH + 255) / 256, 256, 0, stream>>>(W1, W1p, DIN);
    pack_w_kernel<<<(KP * NH + 255) / 256, 256, 0, stream>>>(W2, W2p, NH);
    build_v_kernel<<<(BATCH * KP + 255) / 256, 256, 0, stream>>>(Xi, Xv, cat, num_emb, offsets, Vp);
    fwfm_fs_kernel<<<BATCH / 32, 32, 0, stream>>>(Vp, fwfm_w, fcov, fsb);
    gemm_relu_wmma_kernel<<<640, 256, 0, stream>>>(Vp, W1p, b1, H1, 1);
    gemm_relu_wmma_kernel<<<640, 256, 0, stream>>>(H1, W2p, b2, H2, 1);
    final_kernel<<<(BATCH + 255) / 256, 256, 0, stream>>>(H2, W3, fsb, bias, out);
}